// MPNNModel_1821066133826
// MI455X (gfx1250) — compile-verified
//
#include <hip/hip_runtime.h>

// ---------------------------------------------------------------------------
// MPNN forward for MI455X (gfx1250, wave32). All GEMMs run on
// V_WMMA_F32_16X16X4_F32 (full fp32 accuracy; BN + scatter-max are
// sensitive). h table (12.8MB) lives in the 192MB L2, so edge gathers are
// L2-resident. Edge kernel register-blocks 4 row-tiles per wave so each
// weight (B) fragment feeds 4 WMMAs.
// ---------------------------------------------------------------------------

typedef __attribute__((ext_vector_type(2))) float v2f;
typedef __attribute__((ext_vector_type(8))) float v8f;

#define N_NODES 50000
#define N_EDGES 800000
#define HID 64
#define EDGE_DIM 16
#define MSG_K 144            // 2*HID + EDGE_DIM
#define BN_EPS 1e-5f
#define EPB 64               // edges per block
#define ROW_TILES 4          // 16-row tiles per wave

// Monotonic order-preserving encoding of f32 into u32 so scatter-max can use
// integer atomicMax (commutative => deterministic). enc(-inf) = 0x007FFFFF.
__device__ __forceinline__ unsigned enc_f32(float x) {
  unsigned u = __float_as_uint(x);
  return (u & 0x80000000u) ? ~u : (u | 0x80000000u);
}
__device__ __forceinline__ float dec_f32(unsigned e) {
  unsigned u = (e & 0x80000000u) ? (e ^ 0x80000000u) : ~e;
  return __uint_as_float(u);
}

// ---------------------------------------------------------------------------
// Generic Y[M x N] = act(X[M x K] @ W[K x N] + bias), fp32 WMMA 16x16x4.
// One wave computes one 16x16 tile; blockDim.x = (N/16)*32; grid = M/16.
// ---------------------------------------------------------------------------
template <int KDIM, int NDIM, bool RELU>
__global__ __launch_bounds__(128) void gemm_bias_kernel(
    const float* __restrict__ X, const float* __restrict__ W,
    const float* __restrict__ bias, float* __restrict__ Y) {
  const int tid  = threadIdx.x;
  const int wid  = tid >> 5;
  const int lane = tid & 31;
  const int half = lane >> 4;
  const int lr   = lane & 15;
  const int m0   = blockIdx.x * 16;
  const int n    = wid * 16 + lr;

  v8f c = {};
#pragma unroll
  for (int k0 = 0; k0 < KDIM; k0 += 4) {
    const int kk = k0 + 2 * half;
    v2f a = *(const v2f*)&X[(size_t)(m0 + lr) * KDIM + kk];  // 8B aligned
    v2f b;
    b.x = W[(size_t)kk * NDIM + n];
    b.y = W[(size_t)(kk + 1) * NDIM + n];
    c = __builtin_amdgcn_wmma_f32_16x16x4_f32(false, a, false, b,
                                              (short)0, c, false, false);
  }
  const float bn = bias[n];
#pragma unroll
  for (int r = 0; r < 8; ++r) {
    float v = c[r] + bn;
    if (RELU) v = fmaxf(v, 0.0f);
    Y[(size_t)(m0 + r + 8 * half) * NDIM + n] = v;
  }
}

// ---------------------------------------------------------------------------
// Per-edge MLP + scatter-max.  Block = 64 edges, 128 threads (4 waves).
// Each wave owns one 16-col panel of W1/W2 and 4 row-tiles of edges, so each
// B fragment is reused by 4 WMMAs (4x less weight traffic from L2).
// ---------------------------------------------------------------------------
__global__ __launch_bounds__(128) void edge_mlp_scatter_kernel(
    const float* __restrict__ h, const int* __restrict__ edge_index,
    const float* __restrict__ edge_attr,
    const float* __restrict__ W1, const float* __restrict__ B1,
    const float* __restrict__ W2, const float* __restrict__ B2,
    unsigned* __restrict__ agg_enc) {
  __shared__ float sMsg[EPB][148];  // 144 + pad (592B pitch keeps 16B align)
  __shared__ float sHid[EPB][68];
  __shared__ int   sTgt[EPB];

  const int tid = threadIdx.x;
  const int e0  = blockIdx.x * EPB;
  const int* src_idx = edge_index;             // edge_index[0]
  const int* tgt_idx = edge_index + N_EDGES;   // edge_index[1]

  if (tid < EPB) sTgt[tid] = tgt_idx[e0 + tid];

  // Gather: EPB rows x 36 float4 segments; col = seg*4 (tgt | src | attr).
  for (int i = tid; i < EPB * 36; i += 128) {
    const int row = i / 36;
    const int seg = i - row * 36;
    float4 v;
    if (seg < 16) {
      const int node = tgt_idx[e0 + row];
      v = ((const float4*)(h + (size_t)node * HID))[seg];
    } else if (seg < 32) {
      const int node = src_idx[e0 + row];
      v = ((const float4*)(h + (size_t)node * HID))[seg - 16];
    } else {
      v = ((const float4*)(edge_attr + (size_t)(e0 + row) * EDGE_DIM))[seg - 32];
    }
    *(float4*)&sMsg[row][seg * 4] = v;
  }
  __syncthreads();

  const int wid  = tid >> 5;
  const int lane = tid & 31;
  const int half = lane >> 4;
  const int lr   = lane & 15;
  const int n    = wid * 16 + lr;

  // Layer 1: [EPB x 144] @ [144 x 64] -> relu -> sHid
  v8f acc[ROW_TILES] = {};
#pragma unroll 2
  for (int k0 = 0; k0 < MSG_K; k0 += 4) {
    const int kk = k0 + 2 * half;
    v2f b;
    b.x = W1[(size_t)kk * HID + n];
    b.y = W1[(size_t)(kk + 1) * HID + n];
#pragma unroll
    for (int t = 0; t < ROW_TILES; ++t) {
      v2f a = *(const v2f*)&sMsg[t * 16 + lr][kk];
      acc[t] = __builtin_amdgcn_wmma_f32_16x16x4_f32(false, a, false, b,
                                                     (short)0, acc[t],
                                                     false, false);
    }
  }
  const float bias1 = B1[n];
#pragma unroll
  for (int t = 0; t < ROW_TILES; ++t)
#pragma unroll
    for (int r = 0; r < 8; ++r)
      sHid[t * 16 + r + 8 * half][n] = fmaxf(acc[t][r] + bias1, 0.0f);
  __syncthreads();

  // Layer 2: [EPB x 64] @ [64 x 64] + b2, then scatter-max into agg_enc.
  v8f acc2[ROW_TILES] = {};
#pragma unroll 2
  for (int k0 = 0; k0 < HID; k0 += 4) {
    const int kk = k0 + 2 * half;
    v2f b;
    b.x = W2[(size_t)kk * HID + n];
    b.y = W2[(size_t)(kk + 1) * HID + n];
#pragma unroll
    for (int t = 0; t < ROW_TILES; ++t) {
      v2f a = *(const v2f*)&sHid[t * 16 + lr][kk];
      acc2[t] = __builtin_amdgcn_wmma_f32_16x16x4_f32(false, a, false, b,
                                                      (short)0, acc2[t],
                                                      false, false);
    }
  }
  const float bias2 = B2[n];
#pragma unroll
  for (int t = 0; t < ROW_TILES; ++t)
#pragma unroll
    for (int r = 0; r < 8; ++r) {
      const int row = t * 16 + r + 8 * half;
      atomicMax(&agg_enc[(size_t)sTgt[row] * HID + n],
                enc_f32(acc2[t][r] + bias2));
    }
}

// agg := enc(-inf) everywhere; stats := 0
__global__ void init_kernel(unsigned* __restrict__ agg_enc,
                            float* __restrict__ stats) {
  const size_t total  = (size_t)N_NODES * HID;
  const size_t stride = (size_t)gridDim.x * blockDim.x;
  size_t g = (size_t)blockIdx.x * blockDim.x + threadIdx.x;
  if (g < 128) stats[g] = 0.0f;
  for (size_t i = g; i < total; i += stride) agg_enc[i] = 0x007FFFFFu;
}

// Decode scatter-max result in place (non-finite -> 0, PyG fill semantics),
// accumulate per-channel sum / sumsq into stats[0..63] / stats[64..127].
__global__ __launch_bounds__(256) void decode_stats_kernel(
    unsigned* __restrict__ agg, float* __restrict__ stats) {
  const int tid = threadIdx.x;
  const int c   = tid & 63;
  const int sub = tid >> 6;  // 0..3
  float s = 0.0f, q = 0.0f;
  for (int row = blockIdx.x * 4 + sub; row < N_NODES; row += gridDim.x * 4) {
    const size_t idx = (size_t)row * HID + c;
    float v = dec_f32(agg[idx]);
    const unsigned u = __float_as_uint(v);
    if ((u & 0x7F800000u) == 0x7F800000u) v = 0.0f;  // !isfinite -> 0
    ((float*)agg)[idx] = v;
    s += v;
    q += v * v;
  }
  __shared__ float ss[256], sq[256];
  ss[tid] = s;
  sq[tid] = q;
  __syncthreads();
  if (sub == 0) {
    atomicAdd(&stats[c],      ss[c] + ss[c + 64] + ss[c + 128] + ss[c + 192]);
    atomicAdd(&stats[64 + c], sq[c] + sq[c + 64] + sq[c + 128] + sq[c + 192]);
  }
}

// h = relu(gamma*(agg-mu)*rsqrt(var+eps)+beta), training-mode batch stats.
__global__ void bn_relu_kernel(const float* __restrict__ agg,
                               const float* __restrict__ stats,
                               const float* __restrict__ gamma,
                               const float* __restrict__ beta,
                               float* __restrict__ h) {
  const size_t total  = (size_t)N_NODES * HID;
  const size_t stride = (size_t)gridDim.x * blockDim.x;
  const float invN    = 1.0f / (float)N_NODES;
  for (size_t i = (size_t)blockIdx.x * blockDim.x + threadIdx.x; i < total;
       i += stride) {
    const int c    = (int)(i & 63);
    const float mu  = stats[c] * invN;
    const float var = stats[64 + c] * invN - mu * mu;
    const float v = gamma[c] * (agg[i] - mu) * rsqrtf(var + BN_EPS) + beta[c];
    h[i] = fmaxf(v, 0.0f);
  }
}

extern "C" void kernel_launch(void* const* d_in, const int* in_sizes, int n_in,
                              void* d_out, int out_size, void* d_ws,
                              size_t ws_size, hipStream_t stream) {
  const float* x          = (const float*)d_in[0];
  const int*   edge_index = (const int*)d_in[1];
  const float* edge_attr  = (const float*)d_in[2];
  const float* w_in       = (const float*)d_in[3];
  const float* b_in       = (const float*)d_in[4];
  const float* w1         = (const float*)d_in[5];
  const float* b1         = (const float*)d_in[6];
  const float* w2         = (const float*)d_in[7];
  const float* b2         = (const float*)d_in[8];
  const float* gamma      = (const float*)d_in[9];
  const float* beta       = (const float*)d_in[10];
  const float* w_m1       = (const float*)d_in[11];
  const float* b_m1       = (const float*)d_in[12];
  const float* w_m2       = (const float*)d_in[13];
  const float* b_m2       = (const float*)d_in[14];
  float* out = (float*)d_out;

  char* ws = (char*)d_ws;
  const size_t hbytes = (size_t)N_NODES * HID * sizeof(float);  // 12.8 MB
  float*    h     = (float*)ws;
  unsigned* agg   = (unsigned*)(ws + hbytes);
  float*    stats = (float*)(ws + 2 * hbytes);
  float*    t     = (float*)agg;  // decoder temp reuses agg after the layers

  // h0 = relu(x @ w_in + b_in)
  gemm_bias_kernel<64, 64, true>
      <<<N_NODES / 16, 128, 0, stream>>>(x, w_in, b_in, h);

  for (int l = 0; l < 2; ++l) {
    init_kernel<<<512, 256, 0, stream>>>(agg, stats);
    edge_mlp_scatter_kernel<<<N_EDGES / EPB, 128, 0, stream>>>(
        h, edge_index, edge_attr, w1 + (size_t)l * MSG_K * HID,
        b1 + (size_t)l * HID, w2 + (size_t)l * HID * HID,
        b2 + (size_t)l * HID, agg);
    decode_stats_kernel<<<256, 256, 0, stream>>>(agg, stats);
    bn_relu_kernel<<<512, 256, 0, stream>>>((const float*)agg, stats,
                                            gamma + l * HID, beta + l * HID, h);
  }

  // out = relu(h @ w_m1 + b_m1) @ w_m2 + b_m2
  gemm_bias_kernel<64, 64, true>
      <<<N_NODES / 16, 128, 0, stream>>>(h, w_m1, b_m1, t);
  gemm_bias_kernel<64, 32, false>
      <<<N_NODES / 16, 64, 0, stream>>>(t, w_m2, b_m2, out);
}